// GATConv_86277303042059
// MI455X (gfx1250) — compile-verified
//
#include <hip/hip_runtime.h>
#include <math.h>

typedef __attribute__((ext_vector_type(2))) float v2f;
typedef __attribute__((ext_vector_type(8))) float v8f;

#define NEG_SLOPE 0.2f

// ---------------------------------------------------------------------------
// K1: proj = X @ W   (N x 128) = (N x 128) * (128 x 128), f32 WMMA 16x16x4.
// Block = 256 threads = 8 waves. Block owns a 16-row strip of X (staged in
// LDS); wave w computes the 16x16 output tile at columns [16w, 16w+16).
// ---------------------------------------------------------------------------
__global__ __launch_bounds__(256) void gat_gemm_proj(
    const float* __restrict__ X, const float* __restrict__ W,
    float* __restrict__ proj, int N)
{
    __shared__ float As[16 * 128];           // 8 KB strip of X
    const int row0 = blockIdx.x * 16;
    const int tid  = threadIdx.x;

    // stage 16x128 A-strip into LDS (clamp OOB rows)
    for (int i = tid; i < 16 * 128; i += 256) {
        int r = row0 + (i >> 7);
        if (r >= N) r = N - 1;
        As[i] = X[(size_t)r * 128 + (i & 127)];
    }
    __syncthreads();

    const int wave = tid >> 5;               // column tile 0..7
    const int lane = tid & 31;
    const int m    = lane & 15;              // A row / B,D column-within-tile
    const int half = lane >> 4;              // 0 or 1
    const int col  = wave * 16 + m;          // global output column

    v8f acc = {};
    for (int k0 = 0; k0 < 128; k0 += 4) {
        const int ka = k0 + 2 * half;
        v2f a, b;
        // A 16x4: lane holds row m; VGPR v -> K = k0 + v + 2*half
        a.x = As[m * 128 + ka];
        a.y = As[m * 128 + ka + 1];
        // B 4x16: lane holds col; VGPR v -> K = k0 + v + 2*half
        b.x = W[(size_t)ka * 128 + col];
        b.y = W[(size_t)(ka + 1) * 128 + col];
        acc = __builtin_amdgcn_wmma_f32_16x16x4_f32(
            /*neg_a=*/false, a, /*neg_b=*/false, b,
            /*c_mod=*/(short)0, acc, /*reuse_a=*/false, /*reuse_b=*/false);
    }

    // D 16x16 f32: VGPR i -> row (i + 8*half), col (lane&15) within tile
#pragma unroll
    for (int i = 0; i < 8; ++i) {
        const int r = row0 + i + 8 * half;
        if (r < N) proj[(size_t)r * 128 + col] = acc[i];
    }
}

// ---------------------------------------------------------------------------
// K2: per-node attention logits  sc_src[n,h] = sum_c proj[n,h,c]*a_src[h,c]
// One thread per (node, head).
// ---------------------------------------------------------------------------
__global__ void gat_logits(const float* __restrict__ proj,
                           const float* __restrict__ a_src,
                           const float* __restrict__ a_tgt,
                           float* __restrict__ sc_src,
                           float* __restrict__ sc_tgt, int N)
{
    const int t = blockIdx.x * blockDim.x + threadIdx.x;
    if (t >= N * 4) return;
    const int h = t & 3;
    const int n = t >> 2;
    const float* p  = proj  + (size_t)n * 128 + h * 32;
    const float* as = a_src + h * 32;
    const float* at = a_tgt + h * 32;
    float s0 = 0.f, s1 = 0.f;
#pragma unroll
    for (int c = 0; c < 32; ++c) {
        const float v = p[c];
        s0 += v * as[c];
        s1 += v * at[c];
    }
    sc_src[t] = s0;
    sc_tgt[t] = s1;
}

// ---------------------------------------------------------------------------
// K3: initialize out = 0, segmax = -inf, denom = 0 (graph-replay safe).
// ---------------------------------------------------------------------------
__global__ void gat_init(float* __restrict__ out, float* __restrict__ segmax,
                         float* __restrict__ denom, int nOut, int nSeg)
{
    const int t = blockIdx.x * blockDim.x + threadIdx.x;
    if (t < nOut) out[t] = 0.f;
    if (t < nSeg) { segmax[t] = -__builtin_huge_valf(); denom[t] = 0.f; }
}

__device__ __forceinline__ float leaky(float x) {
    return x > 0.f ? x : NEG_SLOPE * x;
}

__device__ __forceinline__ void edge_ids(int e, int E,
                                         const int* __restrict__ snd,
                                         const int* __restrict__ rcv,
                                         int& s, int& r)
{
    if (e < E) { s = snd[e]; r = rcv[e]; }
    else       { s = r = e - E; }          // self loop
}

// ---------------------------------------------------------------------------
// K4: per-edge segment max of leaky_relu scores (float atomic max).
// One thread per edge, 4 heads via float4 loads.
// ---------------------------------------------------------------------------
__global__ void gat_edge_max(const int* __restrict__ snd, const int* __restrict__ rcv,
                             const float* __restrict__ sc_src,
                             const float* __restrict__ sc_tgt,
                             float* __restrict__ segmax, int E, int N)
{
    const int e = blockIdx.x * blockDim.x + threadIdx.x;
    if (e >= E + N) return;
    int s, r; edge_ids(e, E, snd, rcv, s, r);
    const float4 ss = ((const float4*)sc_src)[s];
    const float4 st = ((const float4*)sc_tgt)[r];
    atomicMax(&segmax[r * 4 + 0], leaky(ss.x + st.x));
    atomicMax(&segmax[r * 4 + 1], leaky(ss.y + st.y));
    atomicMax(&segmax[r * 4 + 2], leaky(ss.z + st.z));
    atomicMax(&segmax[r * 4 + 3], leaky(ss.w + st.w));
}

// ---------------------------------------------------------------------------
// K5: softmax denominator: denom[r,h] += exp(score - segmax[r,h]).
// ---------------------------------------------------------------------------
__global__ void gat_edge_den(const int* __restrict__ snd, const int* __restrict__ rcv,
                             const float* __restrict__ sc_src,
                             const float* __restrict__ sc_tgt,
                             const float* __restrict__ segmax,
                             float* __restrict__ denom, int E, int N)
{
    const int e = blockIdx.x * blockDim.x + threadIdx.x;
    if (e >= E + N) return;
    int s, r; edge_ids(e, E, snd, rcv, s, r);
    const float4 ss = ((const float4*)sc_src)[s];
    const float4 st = ((const float4*)sc_tgt)[r];
    const float4 mx = ((const float4*)segmax)[r];
    atomicAdd(&denom[r * 4 + 0], __expf(leaky(ss.x + st.x) - mx.x));
    atomicAdd(&denom[r * 4 + 1], __expf(leaky(ss.y + st.y) - mx.y));
    atomicAdd(&denom[r * 4 + 2], __expf(leaky(ss.z + st.z) - mx.z));
    atomicAdd(&denom[r * 4 + 3], __expf(leaky(ss.w + st.w) - mx.w));
}

// ---------------------------------------------------------------------------
// K6: weighted message scatter. One wave32 per edge; lane = channel c within
// head; loop over 4 heads; atomic f32 add into out[r].
// ---------------------------------------------------------------------------
__global__ __launch_bounds__(256) void gat_scatter(
    const int* __restrict__ snd, const int* __restrict__ rcv,
    const float* __restrict__ sc_src, const float* __restrict__ sc_tgt,
    const float* __restrict__ segmax, const float* __restrict__ denom,
    const float* __restrict__ proj, float* __restrict__ out, int E, int N)
{
    const int e    = (int)((blockIdx.x * blockDim.x + threadIdx.x) >> 5);
    const int lane = threadIdx.x & 31;
    if (e >= E + N) return;
    int s, r; edge_ids(e, E, snd, rcv, s, r);

    const float4 ss = ((const float4*)sc_src)[s];
    const float4 st = ((const float4*)sc_tgt)[r];
    const float4 mx = ((const float4*)segmax)[r];
    const float4 dn = ((const float4*)denom)[r];

    float att[4];
    att[0] = __expf(leaky(ss.x + st.x) - mx.x) / dn.x;
    att[1] = __expf(leaky(ss.y + st.y) - mx.y) / dn.y;
    att[2] = __expf(leaky(ss.z + st.z) - mx.z) / dn.z;
    att[3] = __expf(leaky(ss.w + st.w) - mx.w) / dn.w;

    const float* ps = proj + (size_t)s * 128;
    float*       po = out  + (size_t)r * 128;
#pragma unroll
    for (int h = 0; h < 4; ++h) {
        atomicAdd(&po[h * 32 + lane], ps[h * 32 + lane] * att[h]);
    }
}

// ---------------------------------------------------------------------------
extern "C" void kernel_launch(void* const* d_in, const int* in_sizes, int n_in,
                              void* d_out, int out_size, void* d_ws, size_t ws_size,
                              hipStream_t stream)
{
    const float* X     = (const float*)d_in[0];   // [N,128]
    const int*   snd   = (const int*)  d_in[1];   // [E]
    const int*   rcv   = (const int*)  d_in[2];   // [E]
    const float* W     = (const float*)d_in[3];   // [128,128]
    const float* a_src = (const float*)d_in[4];   // [4,32]
    const float* a_tgt = (const float*)d_in[5];   // [4,32]
    float*       out   = (float*)d_out;           // [N,128]

    const int N  = in_sizes[0] / 128;
    const int E  = in_sizes[1];
    const int EP = E + N;                         // edges incl. self loops

    // workspace layout (floats): proj | sc_src | sc_tgt | segmax | denom
    float* ws     = (float*)d_ws;
    float* proj   = ws;
    float* sc_src = proj   + (size_t)N * 128;
    float* sc_tgt = sc_src + (size_t)N * 4;
    float* segmax = sc_tgt + (size_t)N * 4;
    float* denom  = segmax + (size_t)N * 4;

    gat_gemm_proj<<<(N + 15) / 16, 256, 0, stream>>>(X, W, proj, N);
    gat_logits<<<(N * 4 + 255) / 256, 256, 0, stream>>>(proj, a_src, a_tgt,
                                                        sc_src, sc_tgt, N);
    gat_init<<<(N * 128 + 255) / 256, 256, 0, stream>>>(out, segmax, denom,
                                                        N * 128, N * 4);
    gat_edge_max<<<(EP + 255) / 256, 256, 0, stream>>>(snd, rcv, sc_src, sc_tgt,
                                                       segmax, E, N);
    gat_edge_den<<<(EP + 255) / 256, 256, 0, stream>>>(snd, rcv, sc_src, sc_tgt,
                                                       segmax, denom, E, N);
    gat_scatter<<<((size_t)EP * 32 + 255) / 256, 256, 0, stream>>>(
        snd, rcv, sc_src, sc_tgt, segmax, denom, proj, out, E, N);
}